// GINCustom_57492432224296
// MI455X (gfx1250) — compile-verified
//
#include <hip/hip_runtime.h>
#include <hip/hip_bf16.h>

#define N_NODES  50000
#define N_EDGES  600000
#define DIMH     128
#define N_GRAPHS 64
#define BN_EPS   1e-5f

typedef __attribute__((ext_vector_type(2))) float v2f;
typedef __attribute__((ext_vector_type(8))) float v8f;

// ---------------------------------------------------------------- utilities
__global__ void copy_f4_k(const float* __restrict__ src, float* __restrict__ dst, int n4) {
    int i = blockIdx.x * blockDim.x + threadIdx.x;
    if (i < n4) ((float4*)dst)[i] = ((const float4*)src)[i];
}

__global__ void zero_f_k(float* __restrict__ p, int n) {
    int i = blockIdx.x * blockDim.x + threadIdx.x;
    if (i < n) p[i] = 0.f;
}

// --------------------------------------------------- scatter-add over edges
// z[dst] += h[src]; one wave (32 lanes) per edge, 4 floats per lane.
__global__ void scatter_add_edges_k(const float* __restrict__ h,
                                    const int* __restrict__ src,
                                    const int* __restrict__ dst,
                                    float* __restrict__ z) {
    int gid  = blockIdx.x * blockDim.x + threadIdx.x;
    int e    = gid >> 5;
    int lane = gid & 31;
    if (e >= N_EDGES) return;
    int s = src[e];
    int d = dst[e];
    const float4 v = *(const float4*)(h + s * DIMH + lane * 4);
    float* zp = z + d * DIMH + lane * 4;
    atomicAdd(zp + 0, v.x);
    atomicAdd(zp + 1, v.y);
    atomicAdd(zp + 2, v.z);
    atomicAdd(zp + 3, v.w);
}

// --------------------------------------------------- WMMA fp32 GEMM (N=128)
// C[M x 128] = A[M x 128] @ W[128 x 128] + bias, optional ReLU.
// 256 threads = 8 waves; wave w covers rows [blk*128 + w*16, +16).
// W staged TRANSPOSED in LDS with pitch 132 dwords:
//   Wt[n*132 + k] = W[k*128 + n]
// so each B fragment half (K=k+2h, k+2h+1 at column n) is one contiguous,
// 8B-aligned ds_load_b64, and banks {4*lm+2*half+k, +1} mod 64 are all
// distinct across the wave (conflict-free). A is software-pipelined one
// K-step ahead so s_wait_loadcnt overlaps a full 8-WMMA burst.
#define WT_PITCH 132

__global__ __launch_bounds__(256)
void gemm_n128_wmma_k(const float* __restrict__ A, const float* __restrict__ W,
                      const float* __restrict__ bias, float* __restrict__ C,
                      int M, int applyRelu) {
    __shared__ float Wt[DIMH * WT_PITCH];   // ~66 KB of the WGP's 320 KB
    int tid = threadIdx.x;
    for (int i = tid; i < DIMH * DIMH; i += 256) {
        int k = i >> 7;          // row of W
        int n = i & 127;         // col of W
        Wt[n * WT_PITCH + k] = W[i];
    }
    __syncthreads();

    int wave = tid >> 5;
    int lane = tid & 31;
    int row0 = blockIdx.x * 128 + wave * 16;
    if (row0 >= M) return;               // wave-uniform: EXEC stays all-1s

    int half = lane >> 4;                // 0: K {k,k+1}; 1: K {k+2,k+3}
    int lm   = lane & 15;

    v8f zacc = {};
    v8f acc[8];
#pragma unroll
    for (int t = 0; t < 8; ++t) acc[t] = zacc;

    const float* Arow = A + (size_t)(row0 + lm) * DIMH;
    __builtin_prefetch(Arow, 0, 0);      // global_prefetch_b8

    // A fragment 16x4: lanes 0-15 hold K=k,k+1; lanes 16-31 hold K=k+2,k+3
    v2f a_cur = *(const v2f*)(Arow + 2 * half);          // k = 0
    for (int k = 0; k < DIMH - 4; k += 4) {
        v2f a_next = *(const v2f*)(Arow + (k + 4) + 2 * half);
#pragma unroll
        for (int t = 0; t < 8; ++t) {
            v2f b = *(const v2f*)(&Wt[(t * 16 + lm) * WT_PITCH + k + 2 * half]);
            acc[t] = __builtin_amdgcn_wmma_f32_16x16x4_f32(
                false, a_cur, false, b, (short)0, acc[t], false, false);
        }
        a_cur = a_next;
    }
    {   // final K-step (k = DIMH-4), no lookahead load
        const int k = DIMH - 4;
#pragma unroll
        for (int t = 0; t < 8; ++t) {
            v2f b = *(const v2f*)(&Wt[(t * 16 + lm) * WT_PITCH + k + 2 * half]);
            acc[t] = __builtin_amdgcn_wmma_f32_16x16x4_f32(
                false, a_cur, false, b, (short)0, acc[t], false, false);
        }
    }

    // Epilogue: C/D layout — VGPR r: lanes0-15 -> row r, lanes16-31 -> row r+8
#pragma unroll
    for (int t = 0; t < 8; ++t) {
        int col = t * 16 + lm;
        float bv = bias[col];
#pragma unroll
        for (int r = 0; r < 8; ++r) {
            int row = row0 + r + 8 * half;
            float v = acc[t][r] + bv;
            if (applyRelu) v = v > 0.f ? v : 0.f;
            C[(size_t)row * DIMH + col] = v;
        }
    }
}

// ------------------------------------------------------- batchnorm (training)
// stats[f] = sum, stats[128+f] = sum of squares (population variance later).
__global__ void bn_stats_k(const float* __restrict__ y, float* __restrict__ stats,
                           int nRows) {
    int f  = threadIdx.x;            // 0..127, blockDim.x == 128
    int r0 = blockIdx.x * 256;
    int rEnd = r0 + 256; if (rEnd > nRows) rEnd = nRows;
    float s = 0.f, s2 = 0.f;
    for (int r = r0; r < rEnd; ++r) {
        float v = y[(size_t)r * DIMH + f];
        s  += v;
        s2 += v * v;
    }
    atomicAdd(&stats[f], s);
    atomicAdd(&stats[DIMH + f], s2);
}

__global__ void bn_relu_apply_k(float* __restrict__ y, const float* __restrict__ stats,
                                const float* __restrict__ gamma,
                                const float* __restrict__ beta, int n4) {
    int i = blockIdx.x * blockDim.x + threadIdx.x;
    if (i >= n4) return;
    float4 v = ((float4*)y)[i];
    int f0 = (i * 4) & (DIMH - 1);
    const float invN = 1.0f / (float)N_NODES;
    float vv[4] = {v.x, v.y, v.z, v.w};
    float out[4];
#pragma unroll
    for (int j = 0; j < 4; ++j) {
        int f = f0 + j;
        float mu  = stats[f] * invN;
        float var = stats[DIMH + f] * invN - mu * mu;
        float t = gamma[f] * (vv[j] - mu) * rsqrtf(var + BN_EPS) + beta[f];
        out[j] = t > 0.f ? t : 0.f;
    }
    ((float4*)y)[i] = make_float4(out[0], out[1], out[2], out[3]);
}

// -------------------------------------------------------------- global pool
__global__ void pool_add_k(const float* __restrict__ h, const int* __restrict__ batch,
                           float* __restrict__ g, int colOff) {
    int gid = blockIdx.x * blockDim.x + threadIdx.x;
    if (gid >= N_NODES * DIMH) return;
    int node = gid >> 7;
    int f    = gid & 127;
    int b    = batch[node];
    atomicAdd(&g[b * 384 + colOff + f], h[gid]);
}

// ---------------------------------------------------------------- final MLP
__global__ void lin1_k(const float* __restrict__ g, const float* __restrict__ w,
                       const float* __restrict__ b, float* __restrict__ out) {
    int gid = blockIdx.x * blockDim.x + threadIdx.x;
    if (gid >= N_GRAPHS * 384) return;
    int r = gid / 384, c = gid - r * 384;
    float acc = b[c];
    const float* grow = g + r * 384;
    for (int k = 0; k < 384; ++k) acc += grow[k] * w[k * 384 + c];
    out[gid] = acc > 0.f ? acc : 0.01f * acc;   // leaky_relu(0.01)
}

__global__ void lin2_k(const float* __restrict__ gl1, const float* __restrict__ w,
                       const float* __restrict__ b, float* __restrict__ out) {
    int r = threadIdx.x;
    if (r >= N_GRAPHS) return;
    float acc = b[0];
    const float* grow = gl1 + r * 384;
    for (int k = 0; k < 384; ++k) acc += grow[k] * w[k];
    out[r] = acc;
}

// --------------------------------------------------------------- host side
static void run_gin_layer(const float* hin, const float* w1, const float* b1,
                          const float* gamma, const float* beta,
                          const float* w2, const float* b2,
                          const int* src, const int* dst,
                          float* zbuf, float* ybuf, float* stats, float* hout,
                          hipStream_t stream) {
    const int n4 = N_NODES * DIMH / 4;
    // z = hin (identity term of GINConv with eps=0)
    copy_f4_k<<<(n4 + 255) / 256, 256, 0, stream>>>(hin, zbuf, n4);
    // z[dst] += hin[src]
    const int sThreads = N_EDGES * 32;
    scatter_add_edges_k<<<(sThreads + 255) / 256, 256, 0, stream>>>(hin, src, dst, zbuf);
    // y = z @ w1 + b1
    gemm_n128_wmma_k<<<(N_NODES + 127) / 128, 256, 0, stream>>>(zbuf, w1, b1, ybuf,
                                                                N_NODES, 0);
    // batchnorm stats + apply + relu (in place on y)
    zero_f_k<<<1, 256, 0, stream>>>(stats, 2 * DIMH);
    bn_stats_k<<<(N_NODES + 255) / 256, 128, 0, stream>>>(ybuf, stats, N_NODES);
    bn_relu_apply_k<<<(n4 + 255) / 256, 256, 0, stream>>>(ybuf, stats, gamma, beta, n4);
    // hout = relu(y @ w2 + b2)
    gemm_n128_wmma_k<<<(N_NODES + 127) / 128, 256, 0, stream>>>(ybuf, w2, b2, hout,
                                                                N_NODES, 1);
}

extern "C" void kernel_launch(void* const* d_in, const int* in_sizes, int n_in,
                              void* d_out, int out_size, void* d_ws, size_t ws_size,
                              hipStream_t stream) {
    const float* x     = (const float*)d_in[0];
    const int*   ei    = (const int*)d_in[1];       // [2, N_EDGES]
    const int*   src   = ei;
    const int*   dst   = ei + N_EDGES;
    const int*   batch = (const int*)d_in[2];

    // params flattened in dict order: conv1{w1,b1,gamma,beta,w2,b2}, conv2, conv3,
    // lin1_w, lin1_b, lin2_w, lin2_b
    const float* conv[3][6];
    for (int l = 0; l < 3; ++l)
        for (int p = 0; p < 6; ++p)
            conv[l][p] = (const float*)d_in[3 + l * 6 + p];
    const float* lin1_w = (const float*)d_in[21];
    const float* lin1_b = (const float*)d_in[22];
    const float* lin2_w = (const float*)d_in[23];
    const float* lin2_b = (const float*)d_in[24];

    // workspace layout (floats)
    float* ws    = (float*)d_ws;
    const size_t NF = (size_t)N_NODES * DIMH;
    float* zbuf  = ws;                 // scatter result
    float* ybuf  = zbuf + NF;          // hidden after GEMM1
    float* hA    = ybuf + NF;          // layer output ping
    float* hB    = hA + NF;            // layer output pong
    float* stats = hB + NF;            // 2*128 floats
    float* g     = stats + 2 * DIMH;   // [64, 384] pooled
    float* gl1   = g + N_GRAPHS * 384; // [64, 384] after lin1

    // zero the pooled-feature accumulator
    zero_f_k<<<(N_GRAPHS * 384 + 255) / 256, 256, 0, stream>>>(g, N_GRAPHS * 384);

    const int poolBlocks = (N_NODES * DIMH + 255) / 256;

    // layer 1: x -> hA, pool into g[:, 0:128]
    run_gin_layer(x, conv[0][0], conv[0][1], conv[0][2], conv[0][3], conv[0][4],
                  conv[0][5], src, dst, zbuf, ybuf, stats, hA, stream);
    pool_add_k<<<poolBlocks, 256, 0, stream>>>(hA, batch, g, 0);

    // layer 2: hA -> hB, pool into g[:, 128:256]
    run_gin_layer(hA, conv[1][0], conv[1][1], conv[1][2], conv[1][3], conv[1][4],
                  conv[1][5], src, dst, zbuf, ybuf, stats, hB, stream);
    pool_add_k<<<poolBlocks, 256, 0, stream>>>(hB, batch, g, 128);

    // layer 3: hB -> hA, pool into g[:, 256:384]
    run_gin_layer(hB, conv[2][0], conv[2][1], conv[2][2], conv[2][3], conv[2][4],
                  conv[2][5], src, dst, zbuf, ybuf, stats, hA, stream);
    pool_add_k<<<poolBlocks, 256, 0, stream>>>(hA, batch, g, 256);

    // readout MLP
    lin1_k<<<(N_GRAPHS * 384 + 255) / 256, 256, 0, stream>>>(g, lin1_w, lin1_b, gl1);
    lin2_k<<<1, 64, 0, stream>>>(gl1, lin2_w, lin2_b, (float*)d_out);
}